// OptNet_LearnD_62311385530567
// MI455X (gfx1250) — compile-verified
//
#include <hip/hip_runtime.h>
#include <math.h>

// OptNet QP (batched IPM) for MI455X / gfx1250.
//  - Q is diagonal (tril(ones)*chol(diag) is elementwise => L diagonal).
//  - G=[[D,-I],[-D,-I]] => Schur-reduce the 127x127 KKT solve to a 64x64 SPD
//    system S = diag(q1) + D^T diag(v) D, solved by blocked Cholesky.
//  - S formation and Cholesky trailing updates use V_WMMA_F32_16X16X4_F32
//    (fp32 WMMA: IPM needs fp32 accuracy; problem is compute bound, ~11 GFLOP
//    total, all data resident in LDS/L2).
//  - One 256-thread workgroup (8 wave32) per batch element, B=1024 blocks.

#define NF 64          // nF
#define NI 63          // nF-1 rows of D / half the inequality count
#define LD 68          // padded row stride of S in LDS (avoid bank conflicts)
#define NITER 15
#define SIGMA 0.1f
#define BIGF 1e9f
#define Q1D (1.0f + 1.0e-4f)     // diag(Q) first 64 entries
#define Q2D (1.0e-8f + 1.0e-4f)  // diag(Q) last 63 entries

typedef __attribute__((ext_vector_type(2))) float v2f;
typedef __attribute__((ext_vector_type(8))) float v8f;

__global__ __launch_bounds__(256) void optnet_ipm(const float* __restrict__ X,
                                                  const float* __restrict__ Dg,
                                                  float* __restrict__ out)
{
    __shared__ float sD[NF * NF];     // D padded to 64x64 (row 63 zero)
    __shared__ float sS[NF * LD];     // Schur matrix / Cholesky factor
    __shared__ float sz1[NF], sz2[NF], ss1[NF], ss2[NF], sl1[NF], sl2[NF];
    __shared__ float sp1[NF];
    __shared__ float se[NF], srd1[NF], srp1[NF], srp2[NF], srhs2[NF];
    __shared__ float swm[NF], sdd[NF], sv[NF];
    __shared__ float sy[NF], sy2[NF], sdz[NF];
    __shared__ float sDinv[NF], sred[NF], sScal[2];

    const int tid  = threadIdx.x;
    const int b    = blockIdx.x;
    const int wave = tid >> 5;
    const int lane = tid & 31;
    const int half = lane >> 4;   // which 16-lane half of the wave
    const int idx  = lane & 15;   // row/col index within a 16-tile

    // ---- load D (shared across the whole batch) and init per-batch state ----
    for (int i = tid; i < NF * NF; i += 256) {
        int r = i >> 6;
        sD[i] = (r < NI) ? Dg[i] : 0.0f;   // zero-pad row 63
    }
    if (tid < NF) {
        sp1[tid] = -X[b * NF + tid];       // p1 = -x_b ; p2 = 13 (constant)
        sz1[tid] = 0.f; sz2[tid] = 0.f;
        ss1[tid] = 1.f; ss2[tid] = 1.f;
        sl1[tid] = 1.f; sl2[tid] = 1.f;
    }

    for (int it = 0; it < NITER; ++it) {
        __syncthreads();
        // ---------- residuals rd, rp and mu partials ----------
        if (tid < NF) {
            float acc = 0.f;                           // D^T (lam1-lam2)
            for (int r = 0; r < NI; ++r) acc += sD[r*NF + tid] * (sl1[r] - sl2[r]);
            srd1[tid] = Q1D * sz1[tid] + sp1[tid] + acc;
        }
        if (tid < NI) {
            float u = 0.f;                             // u = D z1
            for (int j = 0; j < NF; ++j) u += sD[tid*NF + j] * sz1[j];
            srp1[tid] =  u - sz2[tid] + ss1[tid];
            srp2[tid] = -u - sz2[tid] + ss2[tid];
        }
        if (tid < NF) sred[tid] = (tid < NI) ? (ss1[tid]*sl1[tid] + ss2[tid]*sl2[tid]) : 0.f;
        __syncthreads();
        if (tid == 0) {
            float a = 0.f;
            for (int i = 0; i < NF; ++i) a += sred[i];
            sScal[0] = SIGMA * a / 126.0f;             // mu
        }
        __syncthreads();
        const float mu = sScal[0];

        // ---------- barrier weights, Schur diagonal quantities ----------
        if (tid < NI) {
            float s1 = ss1[tid], s2v = ss2[tid], l1 = sl1[tid], l2 = sl2[tid];
            float w1 = l1 / s1, w2 = l2 / s2v;
            float t1 = (mu - s1*l1  + l1*srp1[tid]) / s1;
            float t2 = (mu - s2v*l2 + l2*srp2[tid]) / s2v;
            float wp = w1 + w2, wm = w1 - w2, dd = Q2D + wp;
            swm[tid] = wm; sdd[tid] = dd;
            sv[tid]  = wp - wm*wm/dd;                  // v = wp - wm^2/dd  (>0)
            float rd2v = Q2D*sz2[tid] + 13.0f - (l1 + l2);
            float rhs2 = -(rd2v - (t1 + t2));
            srhs2[tid] = rhs2;
            se[tid]    = wm*rhs2/dd - (t1 - t2);       // combined D^T operand
        }
        if (tid == NI) { sv[NI] = 0.f; se[NI] = 0.f; } // pads
        __syncthreads();

        // ---------- rhs_s = -rd1 + D^T e ; and S = diag(q1)+D^T diag(v) D ----
        if (tid < NF) {
            float acc = 0.f;
            for (int r = 0; r < NI; ++r) acc += sD[r*NF + tid] * se[r];
            sy[tid] = -srd1[tid] + acc;                // rhs for Schur system
        }
        // WMMA formation of S: 4x4 tiles of 16x16, K=64 in steps of 4.
        for (int t = wave; t < 16; t += 8) {
            int TI = t >> 2, TJ = t & 3;
            v8f acc = {0.f,0.f,0.f,0.f,0.f,0.f,0.f,0.f};
            #pragma unroll
            for (int kc = 0; kc < 16; ++kc) {
                int k0 = kc*4 + half*2;
                v2f A, Bf;
                A.x  = sv[k0]     * sD[k0*NF     + TI*16 + idx]; // (diag(v)D)^T
                A.y  = sv[k0 + 1] * sD[(k0+1)*NF + TI*16 + idx];
                Bf.x = sD[k0*NF     + TJ*16 + idx];
                Bf.y = sD[(k0+1)*NF + TJ*16 + idx];
                acc = __builtin_amdgcn_wmma_f32_16x16x4_f32(
                          false, A, false, Bf, (short)0, acc, false, false);
            }
            #pragma unroll
            for (int vv = 0; vv < 8; ++vv) {
                int row = TI*16 + vv + 8*half;
                int col = TJ*16 + idx;
                float val = acc[vv];
                if (row == col) val += Q1D;            // + diag(q1)
                sS[row*LD + col] = val;
            }
        }
        __syncthreads();

        // ---------- blocked Cholesky of S (NB=16), WMMA trailing updates ----
        for (int kb = 0; kb < 4; ++kb) {
            const int off = kb*16;
            if (tid == 0) {                            // 16x16 diag factor
                for (int j = 0; j < 16; ++j) {
                    int jj = off + j;
                    float piv = sqrtf(sS[jj*LD + jj]);
                    sS[jj*LD + jj] = piv;
                    float inv = 1.0f / piv;
                    for (int i = j+1; i < 16; ++i) sS[(off+i)*LD + jj] *= inv;
                    for (int c = j+1; c < 16; ++c) {
                        float lc = sS[(off+c)*LD + jj];
                        for (int i = c; i < 16; ++i)
                            sS[(off+i)*LD + off + c] -= sS[(off+i)*LD + jj] * lc;
                    }
                }
            }
            __syncthreads();
            // panel: L21 = A21 * L11^{-T}, one thread per row
            int r = off + 16 + tid;
            if (r < NF) {
                float xr[16];
                #pragma unroll
                for (int j = 0; j < 16; ++j) {
                    float a = sS[r*LD + off + j];
                    #pragma unroll
                    for (int t2 = 0; t2 < 16; ++t2)
                        if (t2 < j) a -= xr[t2] * sS[(off+j)*LD + off + t2];
                    xr[j] = a / sS[(off+j)*LD + off + j];
                }
                #pragma unroll
                for (int j = 0; j < 16; ++j) sS[r*LD + off + j] = xr[j];
            }
            __syncthreads();
            // trailing: A22 -= L21 L21^T  (K=16 => 4 chained f32 WMMAs/tile)
            const int toff = off + 16;
            const int ntb  = (NF - toff) >> 4;         // 3,2,1,0
            for (int t = wave; t < ntb*ntb; t += 8) {
                int ti = t / ntb, tj = t - ti*ntb;
                int R = toff + ti*16, C = toff + tj*16;
                v8f acc;
                #pragma unroll
                for (int vv = 0; vv < 8; ++vv)
                    acc[vv] = sS[(R + vv + 8*half)*LD + C + idx];
                #pragma unroll
                for (int kc = 0; kc < 4; ++kc) {
                    int k0 = off + kc*4 + half*2;
                    v2f A, Bf;
                    A.x  = -sS[(R+idx)*LD + k0];       // negate A: D = C - A*B
                    A.y  = -sS[(R+idx)*LD + k0 + 1];
                    Bf.x =  sS[(C+idx)*LD + k0];
                    Bf.y =  sS[(C+idx)*LD + k0 + 1];
                    acc = __builtin_amdgcn_wmma_f32_16x16x4_f32(
                              false, A, false, Bf, (short)0, acc, false, false);
                }
                #pragma unroll
                for (int vv = 0; vv < 8; ++vv)
                    sS[(R + vv + 8*half)*LD + C + idx] = acc[vv];
            }
            __syncthreads();
        }
        if (tid < NF) sDinv[tid] = 1.0f / sS[tid*LD + tid];

        // ---------- triangular solves: L y = rhs_s ; L^T dz1 = y ----------
        for (int j = 0; j < NF; ++j) {
            __syncthreads();
            float yj = sy[j] * sDinv[j];               // sy[j] read-only here
            if (tid == j) sy2[j] = yj;
            if (tid > j && tid < NF) sy[tid] -= sS[tid*LD + j] * yj;
        }
        for (int j = NF - 1; j >= 0; --j) {
            __syncthreads();
            float xj = sy2[j] * sDinv[j];
            if (tid == j) sdz[j] = xj;
            if (tid < j) sy2[tid] -= sS[j*LD + tid] * xj;  // L^T via row j of L
        }
        __syncthreads();

        // ---------- recover dz2, ds, dlam; fraction-to-boundary step ----------
        if (tid < NI) {
            float ydz = 0.f;                           // D dz1
            for (int j = 0; j < NF; ++j) ydz += sD[tid*NF + j] * sdz[j];
            float dz2 = (srhs2[tid] + swm[tid]*ydz) / sdd[tid];
            float ds1 = -srp1[tid] - ydz + dz2;
            float ds2 = -srp2[tid] + ydz + dz2;
            float s1 = ss1[tid], s2v = ss2[tid], l1 = sl1[tid], l2 = sl2[tid];
            float dl1 = (mu - s1*l1  - l1*ds1) / s1;
            float dl2 = (mu - s2v*l2 - l2*ds2) / s2v;
            srp1[tid] = ds1; srp2[tid] = ds2;          // reuse as d-step stash
            swm[tid]  = dl1; sdd[tid]  = dl2;
            srhs2[tid] = dz2;
            float rmin = BIGF;
            rmin = fminf(rmin, (ds1 < 0.f) ? (-s1 /ds1) : BIGF);
            rmin = fminf(rmin, (ds2 < 0.f) ? (-s2v/ds2) : BIGF);
            rmin = fminf(rmin, (dl1 < 0.f) ? (-l1 /dl1) : BIGF);
            rmin = fminf(rmin, (dl2 < 0.f) ? (-l2 /dl2) : BIGF);
            sred[tid] = rmin;
        }
        if (tid == NI) sred[NI] = BIGF;
        __syncthreads();
        if (tid == 0) {
            float rm = BIGF;
            for (int i = 0; i < NF; ++i) rm = fminf(rm, sred[i]);
            sScal[1] = fminf(1.0f, 0.99f * rm);        // alpha
        }
        __syncthreads();
        const float alpha = sScal[1];
        if (tid < NF) sz1[tid] += alpha * sdz[tid];
        if (tid < NI) {
            sz2[tid] += alpha * srhs2[tid];
            ss1[tid] += alpha * srp1[tid];
            ss2[tid] += alpha * srp2[tid];
            sl1[tid] += alpha * swm[tid];
            sl2[tid] += alpha * sdd[tid];
        }
    }
    __syncthreads();
    if (tid < NF) out[b*NF + tid] = sz1[tid];          // return z[:, :nF]
}

extern "C" void kernel_launch(void* const* d_in, const int* in_sizes, int n_in,
                              void* d_out, int out_size, void* d_ws, size_t ws_size,
                              hipStream_t stream) {
    const float* x = (const float*)d_in[0];   // [B, 64] f32
    const float* D = (const float*)d_in[1];   // [63, 64] f32
    float* out = (float*)d_out;               // [B, 64] f32
    const int B = in_sizes[0] / NF;           // 1024
    optnet_ipm<<<dim3(B), dim3(256), 0, stream>>>(x, D, out);
    (void)n_in; (void)out_size; (void)d_ws; (void)ws_size;
}